// WaterNetModel_76373108457547
// MI455X (gfx1250) — compile-verified
//
#include <hip/hip_runtime.h>
#include <math.h>

#define NT 2048
#define NS 512
#define NH 64

typedef __attribute__((ext_vector_type(2))) float v2f;
typedef __attribute__((ext_vector_type(8))) float v8f;

__device__ __forceinline__ float wave_sum32(float x) {
  #pragma unroll
  for (int m = 16; m >= 1; m >>= 1) x += __shfl_xor(x, m, 32);
  return x;
}
__device__ __forceinline__ float wave_max32(float x) {
  #pragma unroll
  for (int m = 16; m >= 1; m >>= 1) x = fmaxf(x, __shfl_xor(x, m, 32));
  return x;
}

__global__ __launch_bounds__(32) void waternet_scan(
    const float* __restrict__ P, const float* __restrict__ T,
    const float* __restrict__ w_i, const float* __restrict__ w_o,
    const float* __restrict__ wk, const float* __restrict__ w_s,
    float* __restrict__ out) {
  const int lane = threadIdx.x;   // 0..31, one wave per block
  const int site = blockIdx.x;    // 0..511, wave-uniform
  const int c0 = 2 * lane;

  // ---- one-time weight preprocessing (per lane: channels c0, c0+1) ----
  const float gi0 = 1.f / (1.f + expf(-w_i[c0]));
  const float gi1 = 1.f / (1.f + expf(-w_i[c0 + 1]));
  const float gk0 = 1.f / (1.f + expf(-wk[c0]));
  const float gk1 = 1.f / (1.f + expf(-wk[c0 + 1]));
  const float melt0 = expf(w_s[c0]) + 1.f;
  const float melt1 = expf(w_s[c0 + 1]) + 1.f;
  const float wo0 = w_o[c0], wo1 = w_o[c0 + 1];
  const float mo = wave_max32(fmaxf(wo0, wo1));          // softmax max-sub
  const float e0 = expf(wo0 - mo), e1 = expf(wo1 - mo);
  const float esum = wave_sum32(e0 + e1);
  const float a0 = e0 / esum, a1 = e1 / esum;

  // ---- output pointers (d_out = Q | H | S, flat in return order) ----
  float* Qo = out;                               // NT*NS
  float* Ho = out + (size_t)NT * NS;             // NT*NS*NH
  float* So = Ho + (size_t)NT * NS * NH;         // NT*NS*NH

  float* hp = Ho + (size_t)site * NH + c0;
  float* sp = So + (size_t)site * NH + c0;
  float* qp = Qo + site;
  const size_t hs_stride = (size_t)NS * NH;
  size_t in_off = (size_t)site;

  // WMMA constants: B = all-ones (B[k][n]=1), C = 0
  v2f bones; bones.x = 1.f; bones.y = 1.f;
  v8f cz = {0.f, 0.f, 0.f, 0.f, 0.f, 0.f, 0.f, 0.f};

  float s0 = 0.f, s1 = 0.f, h0 = 0.f, h1 = 0.f;

  #pragma unroll 1
  for (int t = 0; t < NT; ++t) {
    // wave-uniform forcing inputs -> scalar loads
    const float Pk = P[in_off];
    const float Tk = T[in_off];
    in_off += NS;
    const float trel  = fmaxf(Tk, 0.f);
    const float pcold = (Tk < 0.f) ? Pk : 0.f;
    const float prain = (Tk > 0.f) ? Pk : 0.f;

    // SnowBucket + LinearBucket, channel c0
    float sm0 = trel * melt0;
    float m0  = fminf(sm0, s0);
    s0 = (s0 - m0) + pcold;
    float x0   = prain + m0;
    float xin0 = x0 * gi0;
    float t0   = xin0 + h0;
    float q0   = t0 * gk0;
    h0 = (h0 - q0) + xin0;

    // channel c0+1
    float sm1 = trel * melt1;
    float m1  = fminf(sm1, s1);
    s1 = (s1 - m1) + pcold;
    float x1   = prain + m1;
    float xin1 = x1 * gi1;
    float t1   = xin1 + h1;
    float q1   = t1 * gk1;
    h1 = (h1 - q1) + xin1;

    // ---- Q = sum over 64 channels of q*a, via V_WMMA_F32_16X16X4_F32 ----
    // per-lane partial product
    float qa = q0 * a0 + q1 * a1;
    // A layout (16x4 f32): reg0 = A[m,0] (lanes 0-15) / A[m,2] (lanes 16-31),
    // reg1 = A[m,1]/A[m,3].  A = {qa, 0}, B = ones  =>  C[m][n] = qa[m] + qa[m+16]
    v2f aop; aop.x = qa; aop.y = 0.f;
    v8f c = __builtin_amdgcn_wmma_f32_16x16x4_f32(
        /*neg_a=*/false, aop, /*neg_b=*/false, bones,
        /*c_mod=*/(short)0, cz, /*reuse_a=*/false, /*reuse_b=*/false);
    // lanes 0-15: sum over M=0..7 ; lanes 16-31: sum over M=8..15
    float sc = ((c[0] + c[1]) + (c[2] + c[3])) + ((c[4] + c[5]) + (c[6] + c[7]));
    float Qv = sc + __shfl_xor(sc, 16, 32);   // total over all 32 lanes

    // ---- stores: coalesced 8B/lane for H and S, lane0 scalar for Q ----
    *reinterpret_cast<float2*>(hp) = make_float2(h0, h1);
    *reinterpret_cast<float2*>(sp) = make_float2(s0, s1);
    if (lane == 0) *qp = Qv;
    hp += hs_stride;
    sp += hs_stride;
    qp += NS;
  }
}

extern "C" void kernel_launch(void* const* d_in, const int* in_sizes, int n_in,
                              void* d_out, int out_size, void* d_ws, size_t ws_size,
                              hipStream_t stream) {
  const float* P   = (const float*)d_in[0];
  const float* T   = (const float*)d_in[1];
  const float* w_i = (const float*)d_in[2];
  const float* w_o = (const float*)d_in[3];
  // d_in[4] = w_c is dead code in the reference forward pass
  const float* wk  = (const float*)d_in[5];
  const float* w_s = (const float*)d_in[6];
  waternet_scan<<<NS, 32, 0, stream>>>(P, T, w_i, w_o, wk, w_s, (float*)d_out);
}